// YoloX_24936580121225
// MI455X (gfx1250) — compile-verified
//
#include <hip/hip_runtime.h>
#include <stdint.h>

// ---------------- problem constants (match reference) ----------------
#define BATCH      16
#define NCLS       80
#define NCH        85            // 5 + NCLS
#define NANCH      25200         // (80*80 + 40*40 + 20*20) * 3
#define MAXDET     1000
#define CONF_TH    0.05f
#define IOU_TH     0.6f
#define MAX_WH     4096.0f

#define TILE       128                       // anchors per decode workgroup
#define NT         ((NANCH + TILE - 1) / TILE)   // 197 tiles per batch
#define CHUNK      8192                      // bitonic chunk (64KB LDS of u64)
#define NCHUNK     4                         // 4*8192 = 32768 >= 25200

typedef unsigned int u32x4 __attribute__((ext_vector_type(4)));
typedef int          i32x8 __attribute__((ext_vector_type(8)));
typedef int          i32x4 __attribute__((ext_vector_type(4)));

#if defined(__has_builtin)
#if __has_builtin(__builtin_amdgcn_tensor_load_to_lds)
#define HAVE_TDM 1
#endif
#endif
#ifndef HAVE_TDM
#define HAVE_TDM 0
#endif

__device__ __forceinline__ float sigf(float x) { return 1.0f / (1.0f + __expf(-x)); }

// ---------------------------------------------------------------------
// Kernel 1: decode.  One workgroup = 128 anchors of one batch image.
// TDM stages the 85x128 f32 tile (row stride NANCH) into LDS in one DMA.
// ---------------------------------------------------------------------
__global__ void decode_kernel(const float* __restrict__ dt,
                              const float* __restrict__ anchors,
                              float* __restrict__ boxes,
                              float* __restrict__ conf,
                              int*   __restrict__ cid)
{
  const int blk  = blockIdx.x;
  const int b    = blk / NT;
  const int t    = blk % NT;
  const int n0   = t * TILE;
  const int lane = threadIdx.x;
  const int n    = n0 + lane;

#if HAVE_TDM
  __shared__ float tile[NCH * TILE];          // 85*128*4 = 43.5 KB
  if (threadIdx.x < 32) {                     // first wave32 issues the DMA
    unsigned long long ga =
        (unsigned long long)(const void*)(dt + (size_t)b * NCH * NANCH + n0);
    unsigned lds_base = (unsigned)(unsigned long long)(const void*)&tile[0];

    // D# group 0: count=1 | lds_addr | global_addr[56:0] | type=2
    u32x4 g0 = {0u, 0u, 0u, 0u};
    g0[0] = 1u;
    g0[1] = lds_base;
    g0[2] = (unsigned)(ga & 0xFFFFFFFFull);
    g0[3] = (unsigned)((ga >> 32) & 0x01FFFFFFull) | (2u << 30);

    // D# group 1: data_size=4B, 2D tensor (cols remaining x 85 rows),
    // tile = 128 x 85, row stride = NANCH elements. OOB cols read as 0.
    unsigned td0 = (unsigned)(NANCH - n0);    // tensor width from tile start
    unsigned td1 = (unsigned)NCH;             // 85 rows
    i32x8 g1 = {0, 0, 0, 0, 0, 0, 0, 0};
    g1[0] = (int)(2u << 16);                                 // data_size=2 -> 4B
    g1[1] = (int)((td0 & 0xFFFFu) << 16);                    // tensor_dim0[15:0]
    g1[2] = (int)((td0 >> 16) | ((td1 & 0xFFFFu) << 16));    // dim0 hi | dim1 lo
    g1[3] = (int)((td1 >> 16) | ((unsigned)TILE << 16));     // dim1 hi | tile_dim0
    g1[4] = (int)td1;                                        // tile_dim1=85, tile_dim2=0
    g1[5] = (int)(unsigned)NANCH;                            // tensor_dim0_stride lo
    i32x4 gz4 = {0, 0, 0, 0};
    i32x8 gz8 = {0, 0, 0, 0, 0, 0, 0, 0};
    // 6-arg toolchain signature: (g0, g1, g2, g3, g4, cpol)
    __builtin_amdgcn_tensor_load_to_lds(g0, g1, gz4, gz4, gz8, 0);
#if __has_builtin(__builtin_amdgcn_s_wait_tensorcnt)
    __builtin_amdgcn_s_wait_tensorcnt(0);
#else
    asm volatile("s_wait_tensorcnt 0" ::: "memory");
#endif
  }
  __syncthreads();
#define DTV(c) tile[(c) * TILE + lane]
#else
  const float* dptr = dt + (size_t)b * NCH * NANCH + n;
#define DTV(c) dptr[(size_t)(c) * NANCH]
#endif

  if (n < NANCH) {
    float ax = anchors[n * 4 + 0], ay = anchors[n * 4 + 1];
    float aw = anchors[n * 4 + 2], ah = anchors[n * 4 + 3];
    float x = ax + DTV(0) * aw;
    float y = ay + DTV(1) * ah;
    float w = aw * __expf(DTV(2));
    float h = ah * __expf(DTV(3));
    float obj = sigf(DTV(4));
    float best = -1.0f; int bi = 0;
#pragma unroll 4
    for (int c = 0; c < NCLS; ++c) {
      float s = obj * sigf(DTV(5 + c));
      if (s > best) { best = s; bi = c; }   // strict > : first max (jnp.argmax)
    }
    float cf = (best > CONF_TH) ? best : 0.0f;
    size_t o = (size_t)b * NANCH + n;
    boxes[o * 4 + 0] = x - 0.5f * w;
    boxes[o * 4 + 1] = y - 0.5f * h;
    boxes[o * 4 + 2] = x + 0.5f * w;
    boxes[o * 4 + 3] = y + 0.5f * h;
    conf[o] = cf;
    cid[o]  = bi;
  }
#undef DTV
}

// ---------------------------------------------------------------------
// Bitonic sort (descending) of M u64 keys in LDS.  key = conf_bits<<32 | ~idx
// so descending key order == conf descending, ties -> smaller index first,
// exactly matching jax.lax.top_k's stable ordering.
// ---------------------------------------------------------------------
__device__ __forceinline__ void bitonic_desc(unsigned long long* k, int M)
{
  const int tid = threadIdx.x, nth = blockDim.x;
  for (int kk = 2; kk <= M; kk <<= 1) {
    for (int j = kk >> 1; j > 0; j >>= 1) {
      __syncthreads();
      for (int i = tid; i < M; i += nth) {
        int ixj = i ^ j;
        if (ixj > i) {
          unsigned long long a = k[i], c = k[ixj];
          bool descBlk = ((i & kk) == 0);
          if (descBlk ? (a < c) : (a > c)) { k[i] = c; k[ixj] = a; }
        }
      }
    }
  }
  __syncthreads();
}

// Kernel 2: sort each 8192-chunk of one batch; keep its top 1024.
// (global top-1000 is contained in the union of per-chunk top-1024)
__global__ void sort_chunk_kernel(const float* __restrict__ conf,
                                  unsigned long long* __restrict__ ckeys)
{
  __shared__ unsigned long long keys[CHUNK];            // 64 KB LDS
  const int b = blockIdx.x / NCHUNK;
  const int c = blockIdx.x % NCHUNK;
  const int base = c * CHUNK;
  for (int e = threadIdx.x; e < CHUNK; e += blockDim.x) {
    int nn = base + e;
    unsigned long long kk = 0ull;
    if (nn < NANCH) {
      unsigned vb = __float_as_uint(conf[(size_t)b * NANCH + nn]);
      kk = ((unsigned long long)vb << 32) | (unsigned)(~(unsigned)nn);
    }
    keys[e] = kk;
  }
  bitonic_desc(keys, CHUNK);
  for (int e = threadIdx.x; e < 1024; e += blockDim.x)
    ckeys[(size_t)(b * NCHUNK + c) * 1024 + e] = keys[e];
}

// Kernel 3: merge 4x1024 candidates -> exact sorted top-1000, and build the
// detection table det[b][1024][8] = {x1o,y1o,x2o,y2o,area,conf,cid,off}
// (coords offset by cid*MAX_WH, as the reference does for class-aware IoU).
__global__ void topk_det_kernel(const unsigned long long* __restrict__ ckeys,
                                const float* __restrict__ boxes,
                                const int*   __restrict__ cid,
                                float* __restrict__ det)
{
  __shared__ unsigned long long keys[NCHUNK * 1024];    // 32 KB LDS
  const int b = blockIdx.x;
  for (int e = threadIdx.x; e < NCHUNK * 1024; e += blockDim.x)
    keys[e] = ckeys[(size_t)b * NCHUNK * 1024 + e];
  bitonic_desc(keys, NCHUNK * 1024);
  for (int e = threadIdx.x; e < 1024; e += blockDim.x) {
    unsigned long long kk = keys[e];
    float cf = __uint_as_float((unsigned)(kk >> 32));
    unsigned idx = ~(unsigned)kk;
    if (idx >= NANCH) { idx = 0; cf = 0.0f; }           // padded slot
    if (e >= MAXDET) cf = 0.0f;                         // beyond top_k
    size_t o = (size_t)b * NANCH + idx;
    float x1 = boxes[o * 4 + 0], y1 = boxes[o * 4 + 1];
    float x2 = boxes[o * 4 + 2], y2 = boxes[o * 4 + 3];
    int cc = cid[o];
    float off = (float)cc * MAX_WH;
    float* dd = det + ((size_t)(b << 10) + e) * 8;
    dd[0] = x1 + off; dd[1] = y1 + off;
    dd[2] = x2 + off; dd[3] = y2 + off;
    dd[4] = fmaxf(x2 - x1, 0.0f) * fmaxf(y2 - y1, 0.0f);
    dd[5] = cf;
    dd[6] = (float)cc;
    dd[7] = off;
  }
}

// Kernel 4: fully parallel suppression-bit matrix: bit(i,j) = j>i && IoU>thr.
// One 1024-thread block per (batch,row i); wave32 ballot packs 32 bits/word.
__global__ void nms_mask_kernel(const float* __restrict__ det,
                                unsigned* __restrict__ mask)
{
  const int b = blockIdx.x / MAXDET;
  const int i = blockIdx.x % MAXDET;
  const int j = threadIdx.x;                             // 0..1023
  const float* di = det + ((size_t)(b << 10) + i) * 8;   // uniform -> scalar
  const float* dj = det + ((size_t)(b << 10) + j) * 8;
  float xx1 = fmaxf(di[0], dj[0]);
  float yy1 = fmaxf(di[1], dj[1]);
  float xx2 = fminf(di[2], dj[2]);
  float yy2 = fminf(di[3], dj[3]);
  float inter = fmaxf(xx2 - xx1, 0.0f) * fmaxf(yy2 - yy1, 0.0f);
  float iou = inter / (di[4] + dj[4] - inter + 1e-9f);
  bool pred = (j > i) && (iou > IOU_TH);
#if __has_builtin(__builtin_amdgcn_ballot_w32)
  unsigned word = __builtin_amdgcn_ballot_w32(pred);
#else
  unsigned word = (unsigned)__ballot(pred);
#endif
  if ((threadIdx.x & 31) == 0)
    mask[(size_t)blockIdx.x * 32 + (threadIdx.x >> 5)] = word;
}

// Kernel 5: the inherently serial greedy scan — one wave32 per batch.
// Lane w owns suppression word w (columns w*32..w*32+31); per step we only
// need two shfl broadcasts + an OR; mask rows are software-pipelined.
__global__ void nms_scan_kernel(const float* __restrict__ det,
                                const unsigned* __restrict__ mask,
                                unsigned* __restrict__ keep)
{
  const int b = blockIdx.x;
  const int lane = threadIdx.x;                          // 0..31
  unsigned confbits = 0u;
  for (int t = 0; t < 32; ++t) {
    float cf = det[((size_t)(b << 10) + lane * 32 + t) * 8 + 5];
    confbits |= (cf > 0.0f) ? (1u << t) : 0u;
  }
  const unsigned* mb = mask + (size_t)b * MAXDET * 32;
  unsigned removed = 0u;
  unsigned m = mb[lane];                                 // row 0
  for (int i = 0; i < MAXDET; ++i) {
    unsigned mnext = (i + 1 < MAXDET) ? mb[(size_t)(i + 1) * 32 + lane] : 0u;
    int w = i >> 5, bit = i & 31;
    unsigned remw = (unsigned)__shfl((int)removed, w, 32);
    unsigned cbw  = (unsigned)__shfl((int)confbits, w, 32);
    bool alive = (((remw >> bit) & 1u) == 0u) && (((cbw >> bit) & 1u) != 0u);
    if (alive) removed |= m;                             // uniform branch
    m = mnext;
  }
  keep[b * 32 + lane] = confbits & ~removed;
}

// Kernel 6: emit (B, 1000, 6) = [x1,y1,x2,y2,conf,cid] (offset removed), else 0.
__global__ void nms_write_kernel(const float* __restrict__ det,
                                 const unsigned* __restrict__ keep,
                                 float* __restrict__ out)
{
  const int b = blockIdx.x;
  const int j = threadIdx.x;
  if (j < MAXDET) {
    const float* dd = det + ((size_t)(b << 10) + j) * 8;
    bool k = ((keep[b * 32 + (j >> 5)] >> (j & 31)) & 1u) != 0u;
    float off = dd[7];
    size_t base = ((size_t)b * MAXDET + j) * 6;
    out[base + 0] = k ? dd[0] - off : 0.0f;
    out[base + 1] = k ? dd[1] - off : 0.0f;
    out[base + 2] = k ? dd[2] - off : 0.0f;
    out[base + 3] = k ? dd[3] - off : 0.0f;
    out[base + 4] = k ? dd[5] : 0.0f;
    out[base + 5] = k ? dd[6] : 0.0f;
  }
}

// ---------------------------------------------------------------------
extern "C" void kernel_launch(void* const* d_in, const int* in_sizes, int n_in,
                              void* d_out, int out_size, void* d_ws, size_t ws_size,
                              hipStream_t stream)
{
  const float* dt      = (const float*)d_in[0];
  const float* anchors = (const float*)d_in[1];
  float* out = (float*)d_out;
  char*  ws  = (char*)d_ws;

  size_t off = 0;
  float* conf = (float*)(ws + off); off += (size_t)BATCH * NANCH * 4;            // 1.6 MB
  int*   cid  = (int*)  (ws + off); off += (size_t)BATCH * NANCH * 4;            // 1.6 MB
  float* boxes= (float*)(ws + off); off += (size_t)BATCH * NANCH * 16;           // 6.45 MB
  unsigned long long* ckeys =
      (unsigned long long*)(ws + off); off += (size_t)BATCH * NCHUNK * 1024 * 8; // 0.5 MB (8B aligned)
  float* det  = (float*)(ws + off); off += (size_t)BATCH * 1024 * 8 * 4;         // 0.5 MB
  unsigned* mask = (unsigned*)(ws + off); off += (size_t)BATCH * MAXDET * 32 * 4;// 2.0 MB
  unsigned* keep = (unsigned*)(ws + off); off += (size_t)BATCH * 32 * 4;         // ~13 MB total

  decode_kernel   <<<BATCH * NT,      TILE, 0, stream>>>(dt, anchors, boxes, conf, cid);
  sort_chunk_kernel<<<BATCH * NCHUNK, 1024, 0, stream>>>(conf, ckeys);
  topk_det_kernel <<<BATCH,           1024, 0, stream>>>(ckeys, boxes, cid, det);
  nms_mask_kernel <<<BATCH * MAXDET,  1024, 0, stream>>>(det, mask);
  nms_scan_kernel <<<BATCH,             32, 0, stream>>>(det, mask, keep);
  nms_write_kernel<<<BATCH,           1024, 0, stream>>>(det, keep, out);

  (void)in_sizes; (void)n_in; (void)out_size; (void)ws_size;
}